// PMFlow_89601607729185
// MI455X (gfx1250) — compile-verified
//
#include <hip/hip_runtime.h>

// PMFlow: 6-step point-mass flow over 16 fixed centers, B = 4,194,304 particles.
// Compute-bound (~1.2 Kflop/particle vs 16 B/particle): optimize VALU op count.
//   d_in[0] = z        (B*2 floats)
//   d_in[1] = centers  (16*2 floats)
//   d_in[2] = mus      (16 floats)
//   d_out   = z_out    (B*2 floats)

#define PM_STEPS  6
#define PM_DT     0.2f
#define PM_ADV    0.05f
#define PM_EPS    1e-4f
#define PM_NK     16
#define PM_BLOCK  256

typedef int pm_v4i __attribute__((ext_vector_type(4)));
typedef __attribute__((address_space(1))) pm_v4i* pm_gptr_t;  // global (device) AS
typedef __attribute__((address_space(3))) pm_v4i* pm_lptr_t;  // LDS (shared) AS

__device__ __forceinline__ float pm_clamp3(float v) {
    return __builtin_fminf(__builtin_fmaxf(v, -3.0f), 3.0f);
}

__global__ __launch_bounds__(PM_BLOCK)
void pmflow_kernel(const float* __restrict__ z,
                   const float* __restrict__ centers,
                   const float* __restrict__ mus,
                   float* __restrict__ out,
                   int nvec)  // nvec = number of float4 (2-particle) packets
{
    const int tid = blockIdx.x * PM_BLOCK + threadIdx.x;

    // Uniform-address reads -> scalar (SMEM) loads; values live in SGPRs and
    // feed VALU as scalar operands for free.
    float cx[PM_NK], cy[PM_NK], mu[PM_NK];
#pragma unroll
    for (int k = 0; k < PM_NK; ++k) {
        cx[k] = centers[2 * k];
        cy[k] = centers[2 * k + 1];
        mu[k] = mus[k];
    }

    if (tid >= nvec) return;

    const float4* zin = (const float4*)z;
    float4 p;

#if defined(__gfx1250__) && \
    __has_builtin(__builtin_amdgcn_global_load_async_to_lds_b128) && \
    __has_builtin(__builtin_amdgcn_s_wait_asynccnt)
    // CDNA5 async global->LDS path (ASYNCcnt). Each lane stages its own 16 B
    // packet; wave-local wait, no barrier needed (disjoint slots).
    __shared__ float4 stage[PM_BLOCK];
    __builtin_amdgcn_global_load_async_to_lds_b128(
        (pm_gptr_t)(zin + tid),
        (pm_lptr_t)&stage[threadIdx.x],
        0, 0);
    __builtin_amdgcn_s_wait_asynccnt(0);
    p = stage[threadIdx.x];
#else
    p = zin[tid];
#endif

    // Two independent particles per thread -> two independent rsq/FMA chains
    // so TRANS latency hides under VALU co-execution.
    float ax = p.x, ay = p.y;
    float bx = p.z, by = p.w;

#pragma unroll 1   // keep step loop rolled: ~2 KB hot loop body
    for (int s = 0; s < PM_STEPS; ++s) {
        float na = 1.0f, nb = 1.0f;
        float sax = 0.0f, say = 0.0f;
        float sbx = 0.0f, sby = 0.0f;

#pragma unroll
        for (int k = 0; k < PM_NK; ++k) {
            // particle A
            float dxa = ax - cx[k];
            float dya = ay - cy[k];
            float r2a = __builtin_fmaf(dxa, dxa, __builtin_fmaf(dya, dya, PM_EPS));
            float ria = __builtin_amdgcn_rsqf(r2a);        // v_rsq_f32: 1/r
            float ta  = mu[k] * ria;                       // mu/r (reused)
            na += ta;
            float wa  = ta * (ria * ria);                  // mu/r^3
            sax = __builtin_fmaf(wa, dxa, sax);            // sum w*rvec
            say = __builtin_fmaf(wa, dya, say);

            // particle B
            float dxb = bx - cx[k];
            float dyb = by - cy[k];
            float r2b = __builtin_fmaf(dxb, dxb, __builtin_fmaf(dyb, dyb, PM_EPS));
            float rib = __builtin_amdgcn_rsqf(r2b);
            float tb  = mu[k] * rib;
            nb += tb;
            float wb  = tb * (rib * rib);
            sbx = __builtin_fmaf(wb, dxb, sbx);
            sby = __builtin_fmaf(wb, dyb, sby);
        }

        // z += DT * (-S/n + advect);  fold sign and DT into one scale.
        float sa = -PM_DT * __builtin_amdgcn_rcpf(na);     // v_rcp_f32
        float sb = -PM_DT * __builtin_amdgcn_rcpf(nb);
        ax = pm_clamp3(__builtin_fmaf(sax, sa, ax));
        ay = pm_clamp3(__builtin_fmaf(say, sa, ay) + PM_DT * PM_ADV);
        bx = pm_clamp3(__builtin_fmaf(sbx, sb, bx));
        by = pm_clamp3(__builtin_fmaf(sby, sb, by) + PM_DT * PM_ADV);
    }

    float4 o;
    o.x = ax; o.y = ay; o.z = bx; o.w = by;
    ((float4*)out)[tid] = o;   // global_store_b128, coalesced
}

extern "C" void kernel_launch(void* const* d_in, const int* in_sizes, int n_in,
                              void* d_out, int out_size, void* d_ws, size_t ws_size,
                              hipStream_t stream) {
    (void)n_in; (void)out_size; (void)d_ws; (void)ws_size;
    const float* z       = (const float*)d_in[0];
    const float* centers = (const float*)d_in[1];
    const float* mus     = (const float*)d_in[2];
    float* out           = (float*)d_out;

    const int nvec = in_sizes[0] / 4;            // 2 particles (one float4) per thread
    const int grid = (nvec + PM_BLOCK - 1) / PM_BLOCK;
    pmflow_kernel<<<grid, PM_BLOCK, 0, stream>>>(z, centers, mus, out, nvec);
}